// SyntacticGCN_39805756900146
// MI455X (gfx1250) — compile-verified
//
#include <hip/hip_runtime.h>
#include <hip/hip_bf16.h>

typedef __attribute__((ext_vector_type(2))) float v2f;
typedef __attribute__((ext_vector_type(8))) float v8f;

#define DD 128

// ---------------------------------------------------------------- zero init
__global__ void __launch_bounds__(256) syngcn_zero(float4* __restrict__ p, long n4) {
    long i = (long)blockIdx.x * 256 + threadIdx.x;
    if (i < n4) p[i] = make_float4(0.f, 0.f, 0.f, 0.f);
}

// ---------------------------------------------------------------- XV/XG GEMM (WMMA f32 16x16x4)
// One wave computes one branch b, one 16-row block m0: a full 16x128 output
// strip (8 accumulators of 16x16) plus the fused gate dot-product (D->1).
//
// A fragment (16x4 f32): lane L (r=L&15, hi=L>>4) holds X[m0+r][k+2*hi] and
// X[m0+r][k+2*hi+1] in a.x/a.y.  B fragment (4x16): lane L holds
// W[k+2*hi][n0+r], W[k+2*hi+1][n0+r].  C/D (16x16 f32): vgpr v, lane L ->
// row m0+v+8*hi, col n0+r.
__global__ void __launch_bounds__(256) syngcn_xv_xg(
    const float* __restrict__ inp,
    const float* __restrict__ W0, const float* __restrict__ W1,
    const float* __restrict__ W2, const float* __restrict__ W3,
    const float* __restrict__ G0, const float* __restrict__ G1,
    const float* __restrict__ G2, const float* __restrict__ G3,
    float* __restrict__ XV, float* __restrict__ XG, int N)
{
    const int wave   = (blockIdx.x * blockDim.x + threadIdx.x) >> 5;
    const int lane   = threadIdx.x & 31;
    const int mtiles = (N + 15) >> 4;
    if (wave >= 4 * mtiles) return;
    const int b  = wave / mtiles;
    const int m0 = (wave - b * mtiles) << 4;

    const float* __restrict__ W = (b == 0) ? W0 : (b == 1) ? W1 : (b == 2) ? W2 : W3;
    const float* __restrict__ G = (b == 0) ? G0 : (b == 1) ? G1 : (b == 2) ? G2 : G3;

    const int r  = lane & 15;
    const int hi = lane >> 4;

    int rowA = m0 + r;
    if (rowA >= N) rowA = N - 1;                 // clamp tail loads (N%16==0 in practice)
    const float* __restrict__ arow = inp + (size_t)rowA * DD + 2 * hi;

    v8f acc[8] = {};
    float gacc = 0.f;

    for (int k = 0; k < DD; k += 4) {
        v2f a;
        a.x = arow[k];
        a.y = arow[k + 1];
        // fused gate GEMV partial (per-lane, reduced across halves later)
        gacc = __builtin_fmaf(a.x, G[k + 2 * hi],     gacc);
        gacc = __builtin_fmaf(a.y, G[k + 2 * hi + 1], gacc);

        const float* __restrict__ bp = W + (size_t)(k + 2 * hi) * DD + r;
#pragma unroll
        for (int nt = 0; nt < 8; ++nt) {
            v2f bb;
            bb.x = bp[nt * 16];
            bb.y = bp[nt * 16 + DD];
            acc[nt] = __builtin_amdgcn_wmma_f32_16x16x4_f32(
                false, a, false, bb, (short)0, acc[nt], false, false);
        }
    }

    // gate: lane(r,hi=0) has k%4 in {0,1}, lane(r,hi=1) has k%4 in {2,3}
    gacc += __shfl_xor(gacc, 16, 32);
    if (hi == 0 && (m0 + r) < N) XG[(size_t)b * N + m0 + r] = gacc;

    // store the 16x128 strip: row = m0 + v + 8*hi, col = nt*16 + r
    float* __restrict__ obase = XV + ((size_t)b * N + m0 + 8 * hi) * DD + r;
#pragma unroll
    for (int v = 0; v < 8; ++v) {
        const int orow = m0 + v + 8 * hi;
        if (orow < N) {
#pragma unroll
            for (int nt = 0; nt < 8; ++nt)
                obase[(size_t)v * DD + nt * 16] = acc[nt][v];
        }
    }
}

// ---------------------------------------------------------------- per-edge gather + gated scatter
// One wave per edge; 128 floats -> float4 per lane.  Output accumulator
// (51 MB) lives in L2 (192 MB), so the f32 atomics stay on-chip.
__global__ void __launch_bounds__(256) syngcn_edges(
    const float* __restrict__ XV, const float* __restrict__ XG,
    const float* __restrict__ b_in,  const float* __restrict__ b_out,
    const float* __restrict__ bg_in, const float* __restrict__ bg_out,
    const int* __restrict__ deprel, const int* __restrict__ deparc,
    const int* __restrict__ esrc,   const int* __restrict__ etgt,
    float* __restrict__ out, int N, int E)
{
    const int e    = (blockIdx.x * blockDim.x + threadIdx.x) >> 5;
    const int lane = threadIdx.x & 31;
    if (e >= E) return;

    const int t   = deparc[e];
    const int src = esrc[e];
    const int tgt = etgt[e];
    const int rel = deprel[e];

    float z = XG[(size_t)t * N + src];
    if (t == 0)      z += bg_in[rel];
    else if (t == 1) z += bg_out[rel];
    const float gate = 1.0f / (1.0f + __expf(-z));

    const float4 m = *((const float4*)(XV + ((size_t)t * N + src) * DD) + lane);
    float4 bb = make_float4(0.f, 0.f, 0.f, 0.f);
    if (t == 0)      bb = *((const float4*)(b_in  + (size_t)rel * DD) + lane);
    else if (t == 1) bb = *((const float4*)(b_out + (size_t)rel * DD) + lane);

    float* o = out + (size_t)tgt * DD + lane * 4;
    unsafeAtomicAdd(o + 0, gate * (m.x + bb.x));
    unsafeAtomicAdd(o + 1, gate * (m.y + bb.y));
    unsafeAtomicAdd(o + 2, gate * (m.z + bb.z));
    unsafeAtomicAdd(o + 3, gate * (m.w + bb.w));
}

// ---------------------------------------------------------------- residual + relu
__global__ void __launch_bounds__(256) syngcn_final(
    float4* __restrict__ out, const float4* __restrict__ inp, long n4)
{
    long i = (long)blockIdx.x * 256 + threadIdx.x;
    if (i < n4) {
        float4 a = out[i];
        float4 x = inp[i];
        float4 r;
        r.x = fmaxf(a.x + x.x, 0.f);
        r.y = fmaxf(a.y + x.y, 0.f);
        r.z = fmaxf(a.z + x.z, 0.f);
        r.w = fmaxf(a.w + x.w, 0.f);
        out[i] = r;
    }
}

extern "C" void kernel_launch(void* const* d_in, const int* in_sizes, int n_in,
                              void* d_out, int out_size, void* d_ws, size_t ws_size,
                              hipStream_t stream) {
    const float* inp      = (const float*)d_in[0];
    const int*   deprel   = (const int*)  d_in[1];
    const int*   deparc   = (const int*)  d_in[2];
    const int*   eidx     = (const int*)  d_in[3];
    const float* V_in     = (const float*)d_in[4];
    const float* b_in     = (const float*)d_in[5];
    const float* V_in_g   = (const float*)d_in[6];
    const float* b_in_g   = (const float*)d_in[7];
    const float* V_out    = (const float*)d_in[8];
    const float* b_out    = (const float*)d_in[9];
    const float* V_out_g  = (const float*)d_in[10];
    const float* b_out_g  = (const float*)d_in[11];
    const float* W_self   = (const float*)d_in[12];
    const float* W_self_g = (const float*)d_in[13];
    const float* W_norel  = (const float*)d_in[14];
    const float* W_norel_g= (const float*)d_in[15];

    const int N = in_sizes[0] / DD;
    const int E = in_sizes[1];

    float* out = (float*)d_out;
    float* XV  = (float*)d_ws;                       // 4*N*128 f32
    float* XG  = XV + (size_t)4 * N * DD;            // 4*N f32

    const long n4 = (long)N * (DD / 4);

    // 1) zero the aggregation buffer (harness poisons d_out)
    syngcn_zero<<<(int)((n4 + 255) / 256), 256, 0, stream>>>((float4*)out, n4);

    // 2) per-node transforms + fused gate dots via f32 WMMA
    const int mtiles = (N + 15) / 16;
    const int waves  = 4 * mtiles;
    syngcn_xv_xg<<<(waves + 7) / 8, 256, 0, stream>>>(
        inp, V_in, V_out, W_self, W_norel,
        V_in_g, V_out_g, W_self_g, W_norel_g, XV, XG, N);

    // 3) per-edge gather + gated atomic scatter
    syngcn_edges<<<(E + 7) / 8, 256, 0, stream>>>(
        XV, XG, b_in, b_out, b_in_g, b_out_g,
        deprel, deparc, eidx, eidx + E, out, N, E);

    // 4) residual + relu
    syngcn_final<<<(int)((n4 + 255) / 256), 256, 0, stream>>>(
        (float4*)out, (const float4*)inp, n4);
}